// ClsHeader_10831907521131
// MI455X (gfx1250) — compile-verified
//
#include <hip/hip_runtime.h>
#include <hip/hip_bf16.h>

typedef __attribute__((ext_vector_type(2))) float v2f;
typedef __attribute__((ext_vector_type(4))) float v4f;
typedef __attribute__((ext_vector_type(8))) float v8f;

#define BN_EPS 1e-5f
#define C_IN   128   // channels per depth level
#define K_DIM  384   // 3 * 128 concat
#define N_DIM  1024
#define M_DIM  512   // B * 64 depth-2 nodes
#define NCLS   40
#define NSAMP  8

static __device__ __forceinline__ v4f vmax4(v4f a, v4f b) {
    v4f r;
    r.x = fmaxf(a.x, b.x); r.y = fmaxf(a.y, b.y);
    r.z = fmaxf(a.z, b.z); r.w = fmaxf(a.w, b.w);
    return r;
}

// ---------------------------------------------------------------------------
// Stage 1: octree max-pool.  out[r, colOffset + c] = max over rowsPerOut
// consecutive input rows of in[., c].  One block per output row.
// blockDim = 256: tid%32 selects a float4 channel group (covers 128 ch),
// tid/32 selects one of 8 row segments.  A wave reads a full contiguous
// 512-byte channel row per iteration -> perfectly coalesced streaming.
// The 153 MB input stream has zero reuse: use non-temporal loads so it
// doesn't evict conv_w / X from the 192 MB L2 before stage 2 runs.
// ---------------------------------------------------------------------------
__global__ void pool_max_kernel(const float* __restrict__ in,
                                float* __restrict__ X,
                                int rowsPerOut, int colOffset) {
    const int tid  = threadIdx.x;
    const int c4   = tid & 31;   // float4 group: channels 4*c4 .. 4*c4+3
    const int rseg = tid >> 5;   // 0..7
    const int r    = blockIdx.x; // output row 0..511
    const int rowsPerSeg = rowsPerOut >> 3;  // 512->64, 64->8, 8->1

    const v4f* base = (const v4f*)(in
        + ((size_t)r * rowsPerOut + (size_t)rseg * rowsPerSeg) * C_IN + c4 * 4);
    v4f m = __builtin_nontemporal_load(base);
    for (int i = 1; i < rowsPerSeg; ++i) {
        v4f v = __builtin_nontemporal_load(base + (size_t)i * (C_IN / 4));
        m = vmax4(m, v);
    }

    __shared__ v4f smem[8][32];
    smem[rseg][c4] = m;
    __syncthreads();
    if (rseg == 0) {
        v4f mm = smem[0][c4];
        #pragma unroll
        for (int s = 1; s < 8; ++s) mm = vmax4(mm, smem[s][c4]);
        *(v4f*)(X + (size_t)r * K_DIM + colOffset + c4 * 4) = mm;   // regular TH: reused by GEMM
    }
}

// ---------------------------------------------------------------------------
// Stage 2: fused GEMM [512,384]x[384,1024] via V_WMMA_F32_16X16X4_F32,
// + BatchNorm(inference) + ReLU + mean over each sample's 64 rows.
// One wave per (sample b, 16-column tile n0).  512 waves total.
// fp32 WMMA operand layouts (ISA 7.12.2):
//   A 16x4 : lane L holds M = L%16, VGPR0/1 = K = koff, koff+1 (koff = 2*(L/16))
//   B  4x16: lane L holds N = L%16, VGPR0/1 = K = koff, koff+1
//   D 16x16: VGPR j -> row (j + 8*(L/16)), col = L%16
// Deterministic: no atomics; lane L and L+16 partial column sums are combined
// with a wave32 shfl_xor(16).
// ---------------------------------------------------------------------------
__global__ void gemm_bn_pool_kernel(const float* __restrict__ X,     // [512,384]
                                    const float* __restrict__ W,     // [384,1024]
                                    const float* __restrict__ gamma,
                                    const float* __restrict__ beta,
                                    const float* __restrict__ mean,
                                    const float* __restrict__ var,
                                    float* __restrict__ P) {         // [8,1024]
    const int lane = threadIdx.x & 31;
    const int wid  = threadIdx.x >> 5;
    const int w    = blockIdx.x * 8 + wid;   // 0..511
    const int b    = w >> 6;                 // sample 0..7
    const int n0   = (w & 63) * 16;          // column tile base
    const int nl   = lane & 15;              // lane's column / A-row index
    const int n    = n0 + nl;
    const int koff = (lane >> 4) * 2;        // 0 or 2

    // Per-column BN affine: y = d*inv + shift
    const float inv   = gamma[n] * rsqrtf(var[n] + BN_EPS);
    const float shift = beta[n] - mean[n] * inv;

    float colsum = 0.0f;
    #pragma unroll
    for (int mt = 0; mt < 4; ++mt) {
        const int m0 = b * 64 + mt * 16;
        const float* arow = X + (size_t)(m0 + nl) * K_DIM + koff;
        v8f acc = {};
        for (int k = 0; k < K_DIM; k += 4) {
            v2f a = *(const v2f*)(arow + k);                    // 8B-aligned (k+koff even)
            const float* wp = W + (size_t)(k + koff) * N_DIM + n;
            v2f bv;
            bv.x = wp[0];
            bv.y = wp[N_DIM];
            acc = __builtin_amdgcn_wmma_f32_16x16x4_f32(
                /*neg_a=*/false, a, /*neg_b=*/false, bv,
                /*c_mod=*/(short)0, acc, /*reuse_a=*/false, /*reuse_b=*/false);
        }
        // BN + ReLU + sum the 8 rows this lane holds (all at column n)
        float s = 0.0f;
        #pragma unroll
        for (int j = 0; j < 8; ++j) {
            float vv = fmaxf(acc[j] * inv + shift, 0.0f);
            s += vv;
        }
        colsum += s;
    }
    // Combine rows 0..7 (lanes 0-15) with rows 8..15 (lanes 16-31)
    colsum += __shfl_xor(colsum, 16, 32);
    if (lane < 16) {
        P[b * N_DIM + n] = colsum * (1.0f / 64.0f);
    }
}

// ---------------------------------------------------------------------------
// Stage 3: head GEMM [8,1024] @ [1024,40] + bias -> logits [8,40]
// ---------------------------------------------------------------------------
__global__ void head_kernel(const float* __restrict__ P,
                            const float* __restrict__ Wh,  // [1024,40]
                            const float* __restrict__ bh,  // [40]
                            float* __restrict__ out) {     // [8,40]
    const int b = blockIdx.x;      // 0..7
    const int j = threadIdx.x;     // 0..63
    if (j >= NCLS) return;
    const float* p = P + (size_t)b * N_DIM;
    float s = bh[j];
    for (int k = 0; k < N_DIM; ++k) {
        s = fmaf(p[k], Wh[(size_t)k * NCLS + j], s);
    }
    out[b * NCLS + j] = s;
}

// ---------------------------------------------------------------------------
extern "C" void kernel_launch(void* const* d_in, const int* in_sizes, int n_in,
                              void* d_out, int out_size, void* d_ws, size_t ws_size,
                              hipStream_t stream) {
    const float* data0  = (const float*)d_in[0];  // [B*8^5, 128]
    const float* data1  = (const float*)d_in[1];  // [B*8^4, 128]
    const float* data2  = (const float*)d_in[2];  // [B*8^3, 128]
    const float* conv_w = (const float*)d_in[3];  // [384, 1024]
    const float* gamma  = (const float*)d_in[4];
    const float* beta   = (const float*)d_in[5];
    const float* mean   = (const float*)d_in[6];
    const float* var    = (const float*)d_in[7];
    const float* head_w = (const float*)d_in[8];  // [1024, 40]
    const float* head_b = (const float*)d_in[9];  // [40]
    float* out = (float*)d_out;

    // Workspace layout
    float* X = (float*)d_ws;               // [512, 384] pooled concat
    float* P = X + (size_t)M_DIM * K_DIM;  // [8, 1024] per-sample mean

    // Stage 1: pool each depth down to 64 nodes/sample, write concat columns
    pool_max_kernel<<<M_DIM, 256, 0, stream>>>(data0, X, 512, 0);
    pool_max_kernel<<<M_DIM, 256, 0, stream>>>(data1, X, 64, 128);
    pool_max_kernel<<<M_DIM, 256, 0, stream>>>(data2, X, 8, 256);

    // Stage 2: WMMA GEMM + BN + ReLU + per-sample mean (512 waves)
    gemm_bn_pool_kernel<<<64, 256, 0, stream>>>(X, conv_w, gamma, beta, mean, var, P);

    // Stage 3: classification head
    head_kernel<<<NSAMP, 64, 0, stream>>>(P, head_w, head_b, out);
}